// GCN2_50096498540829
// MI455X (gfx1250) — compile-verified
//
#include <hip/hip_runtime.h>
#include <math.h>

typedef __attribute__((ext_vector_type(2))) float v2f;
typedef __attribute__((ext_vector_type(8))) float v8f;

#define NNODES 100000
#define NEDGES 3200000
#define IN_C   256
#define HID_C  256
#define OUT_C  64
#define NTILES (NNODES / 16)   // 6250, exact

// ---------------------------------------------------------------- zero fill
__global__ void zero_f32(float* __restrict__ p, long long n) {
  long long i = (long long)blockIdx.x * blockDim.x + threadIdx.x;
  long long stride = (long long)gridDim.x * blockDim.x;
  for (; i < n; i += stride) p[i] = 0.0f;
}

// ---------------------------------------------------------------- GEMM1
// H[N,256] = X[N,256] @ W1^T[256,256] + b1
// One wave per 16-row tile, covering all 16 column tiles (128 accum VGPRs).
// X streamed from HBM exactly once; W1 (256KB) is L2-resident.
__global__ __launch_bounds__(128) void gemm1(const float* __restrict__ X,
                                             const float* __restrict__ W,
                                             const float* __restrict__ bias,
                                             float* __restrict__ H) {
  const int wave = threadIdx.x >> 5;
  const int lane = threadIdx.x & 31;
  const int li = lane & 15;      // M (for A) / N (for B,C,D)
  const int hi = lane >> 4;      // K-pair selector / M-half selector
  const int tile = blockIdx.x * 4 + wave;
  if (tile >= NTILES) return;
  const int row0 = tile * 16;

  const float* pA = X + (long long)(row0 + li) * IN_C + 2 * hi;

  v8f acc[16];
#pragma unroll
  for (int t = 0; t < 16; ++t) {
    float bv = bias[t * 16 + li];
#pragma unroll
    for (int r = 0; r < 8; ++r) acc[t][r] = bv;
  }

#pragma unroll 4
  for (int k = 0; k < IN_C; k += 4) {
    v2f a = *(const v2f*)(pA + k);
#pragma unroll
    for (int t = 0; t < 16; ++t) {
      v2f b = *(const v2f*)(W + (long long)(t * 16 + li) * IN_C + 2 * hi + k);
      acc[t] = __builtin_amdgcn_wmma_f32_16x16x4_f32(
          false, a, false, b, (short)0, acc[t], false, false);
    }
  }

#pragma unroll
  for (int t = 0; t < 16; ++t) {
#pragma unroll
    for (int r = 0; r < 8; ++r) {
      H[(long long)(row0 + r + 8 * hi) * HID_C + t * 16 + li] = acc[t][r];
    }
  }
}

// ---------------------------------------------------------------- GEMM2
// H2[N,64] = relu(AGG[N,256]) @ W2^T[256,64] + b2   (ReLU + bias fused)
__global__ __launch_bounds__(256) void gemm2(const float* __restrict__ A,
                                             const float* __restrict__ W,
                                             const float* __restrict__ bias,
                                             float* __restrict__ H) {
  const int wave = threadIdx.x >> 5;
  const int lane = threadIdx.x & 31;
  const int li = lane & 15;
  const int hi = lane >> 4;
  const int tile = blockIdx.x * 8 + wave;
  if (tile >= NTILES) return;
  const int row0 = tile * 16;

  const float* pA = A + (long long)(row0 + li) * HID_C + 2 * hi;

  v8f acc[4];
#pragma unroll
  for (int t = 0; t < 4; ++t) {
    float bv = bias[t * 16 + li];
#pragma unroll
    for (int r = 0; r < 8; ++r) acc[t][r] = bv;
  }

#pragma unroll 4
  for (int k = 0; k < HID_C; k += 4) {
    v2f a = *(const v2f*)(pA + k);
    a[0] = fmaxf(a[0], 0.0f);          // fused ReLU
    a[1] = fmaxf(a[1], 0.0f);
#pragma unroll
    for (int t = 0; t < 4; ++t) {
      v2f b = *(const v2f*)(W + (long long)(t * 16 + li) * HID_C + 2 * hi + k);
      acc[t] = __builtin_amdgcn_wmma_f32_16x16x4_f32(
          false, a, false, b, (short)0, acc[t], false, false);
    }
  }

#pragma unroll
  for (int t = 0; t < 4; ++t) {
#pragma unroll
    for (int r = 0; r < 8; ++r) {
      H[(long long)(row0 + r + 8 * hi) * OUT_C + t * 16 + li] = acc[t][r];
    }
  }
}

// ---------------------------------------------------------------- edge scatter
// agg[dst] += h[src] for each edge; one block per edge, coalesced.
__global__ __launch_bounds__(256) void spmm_h(const float* __restrict__ h,
                                              const long long* __restrict__ ei,
                                              float* __restrict__ agg) {
  const int e = blockIdx.x;
  const int src = (int)ei[e];
  const int dst = (int)ei[(long long)NEDGES + e];
  float v = h[(long long)src * HID_C + threadIdx.x];
  unsafeAtomicAdd(agg + (long long)dst * HID_C + threadIdx.x, v);
}

__global__ __launch_bounds__(64) void spmm_o(const float* __restrict__ h,
                                             const long long* __restrict__ ei,
                                             float* __restrict__ agg) {
  const int e = blockIdx.x;
  const int src = (int)ei[e];
  const int dst = (int)ei[(long long)NEDGES + e];
  float v = h[(long long)src * OUT_C + threadIdx.x];
  unsafeAtomicAdd(agg + (long long)dst * OUT_C + threadIdx.x, v);
}

// ---------------------------------------------------------------- log-softmax
// In-place over rows of 64; one wave32 per row, 2 channels per lane.
__global__ __launch_bounds__(256) void logsoftmax64(float* __restrict__ out) {
  const int wave = threadIdx.x >> 5;
  const int lane = threadIdx.x & 31;
  const long long row = (long long)blockIdx.x * 8 + wave;
  if (row >= NNODES) return;
  float* p = out + row * OUT_C;
  float a0 = p[lane];
  float a1 = p[lane + 32];
  float m = fmaxf(a0, a1);
#pragma unroll
  for (int o = 16; o; o >>= 1) m = fmaxf(m, __shfl_xor(m, o, 32));
  float s = expf(a0 - m) + expf(a1 - m);
#pragma unroll
  for (int o = 16; o; o >>= 1) s += __shfl_xor(s, o, 32);
  float l = logf(s);
  p[lane] = a0 - m - l;
  p[lane + 32] = a1 - m - l;
}

// ---------------------------------------------------------------- launcher
extern "C" void kernel_launch(void* const* d_in, const int* in_sizes, int n_in,
                              void* d_out, int out_size, void* d_ws, size_t ws_size,
                              hipStream_t stream) {
  const float*     x  = (const float*)d_in[0];
  const long long* ei = (const long long*)d_in[1];   // int64 edge_index [2,E]
  const float*     W1 = (const float*)d_in[2];
  const float*     b1 = (const float*)d_in[3];
  const float*     W2 = (const float*)d_in[4];
  const float*     b2 = (const float*)d_in[5];
  float* out = (float*)d_out;

  // Workspace layout: [h1 | agg1]; h2 reuses h1's buffer after agg1 is built.
  float* h1   = (float*)d_ws;
  float* agg1 = h1 + (long long)NNODES * HID_C;
  float* h2   = h1;  // 25.6MB fits inside 102.4MB h1 region; h1 dead by then

  const long long n_agg1 = (long long)NNODES * HID_C;
  const long long n_out  = (long long)NNODES * OUT_C;

  zero_f32<<<4096, 256, 0, stream>>>(agg1, n_agg1);
  zero_f32<<<4096, 256, 0, stream>>>(out, n_out);

  gemm1<<<(NTILES + 3) / 4, 128, 0, stream>>>(x, W1, b1, h1);
  spmm_h<<<NEDGES, 256, 0, stream>>>(h1, ei, agg1);
  gemm2<<<(NTILES + 7) / 8, 256, 0, stream>>>(agg1, W2, b2, h2);
  spmm_o<<<NEDGES, 64, 0, stream>>>(h2, ei, out);
  logsoftmax64<<<(NNODES + 7) / 8, 256, 0, stream>>>(out);
}